// GCNLayer_55499567399492
// MI455X (gfx1250) — compile-verified
//
#include <hip/hip_runtime.h>

#define D 128          // D_IN == D_OUT == 128
#define LDSW 132       // padded LDS row stride (floats) to avoid bank conflicts

typedef float v2f __attribute__((ext_vector_type(2)));
typedef float v8f __attribute__((ext_vector_type(8)));

// ---------------------------------------------------------------- kernel 1
__global__ void gcn_zero(float* __restrict__ out, float* __restrict__ deg,
                         int n_out, int n_deg) {
    int i = blockIdx.x * blockDim.x + threadIdx.x;
    int stride = gridDim.x * blockDim.x;
    for (int j = i; j < n_out; j += stride) out[j] = 0.0f;
    for (int j = i; j < n_deg; j += stride) deg[j] = 0.0f;
}

// ---------------------------------------------------------------- kernel 2
// One thread per (edge, 4-float chunk): 32 threads cover one edge's 128 feats.
// feature (51 MB) and agg (51 MB) both fit in the 192 MB L2, so this is an
// L2-atomic problem: unsafeAtomicAdd -> native global_atomic_add_f32.
__global__ void gcn_scatter(const float* __restrict__ feat,
                            const int* __restrict__ src,
                            const int* __restrict__ dst,
                            float* __restrict__ agg,
                            float* __restrict__ deg, int E) {
    long long tid = (long long)blockIdx.x * blockDim.x + threadIdx.x;
    int e = (int)(tid >> 5);
    int c = (int)(tid & 31);
    if (e >= E) return;
    int s = src[e];
    int d = dst[e];
    const float4* fin = (const float4*)(feat + (long long)s * D);
    float4 v = fin[c];
    float* ao = agg + (long long)d * D + c * 4;
    unsafeAtomicAdd(ao + 0, v.x);
    unsafeAtomicAdd(ao + 1, v.y);
    unsafeAtomicAdd(ao + 2, v.z);
    unsafeAtomicAdd(ao + 3, v.w);
    if (c == 0) unsafeAtomicAdd(deg + d, 1.0f);
}

// ---------------------------------------------------------------- kernel 3
// 256 threads = 8 waves per block; block owns a 16-node tile.
// LDS-staged h (mean applied), then each wave does a 16x16 output slab via
// 32x chained V_WMMA_F32_16X16X4_F32. In-place update of `out` is safe:
// full tile is in LDS before any store.
__global__ void __launch_bounds__(256)
gcn_gemm(const float* __restrict__ Wm,    // [D_OUT][D_IN] row-major
         const float* __restrict__ bias,  // [D_OUT]
         const float* __restrict__ deg,   // [N]
         float* __restrict__ out,         // in: agg, out: h@W^T+b
         int N) {
    __shared__ __align__(16) float hs[16 * LDSW];

    const int tile = blockIdx.x;
    const int t = threadIdx.x;

    // Stage 16x128 h-tile into LDS with mean normalization.
    for (int i = t; i < 16 * D; i += 256) {
        int row = i >> 7;          // /128
        int col = i & (D - 1);
        int node = tile * 16 + row;
        float v = 0.0f;
        if (node < N) {
            float dg = deg[node];
            if (dg > 0.0f) v = out[(long long)node * D + col] / dg;
        }
        hs[row * LDSW + col] = v;
    }
    __syncthreads();

    const int wave = t >> 5;
    const int lane = t & 31;
    const int m    = lane & 15;        // A row / output N within slab
    const int n    = wave * 16 + m;    // output column (0..127)
    const int koff = (lane < 16) ? 0 : 2;

    const float* hrow = hs + m * LDSW;            // A: h[m][*]
    const float* wrow = Wm + (long long)n * D;    // B[k][n] = W[n][k]

    v8f acc = {};
    #pragma unroll 4
    for (int k0 = 0; k0 < D; k0 += 4) {
        v2f a = *(const v2f*)(hrow + k0 + koff);  // A[m][k0+koff], A[m][k0+koff+1]
        v2f b = *(const v2f*)(wrow + k0 + koff);  // B[k0+koff][n], B[k0+koff+1][n]
        acc = __builtin_amdgcn_wmma_f32_16x16x4_f32(
            /*neg_a=*/false, a, /*neg_b=*/false, b,
            /*c_mod=*/(short)0, acc, /*reuse_a=*/false, /*reuse_b=*/false);
    }

    const float bcol = bias[n];
    const int rbase = (lane < 16) ? 0 : 8;
    // C/D layout: vgpr r -> row rbase+r, col n. Base pointer once; stores at
    // immediate offsets r*D*4. Whole-tile-in-range test is uniform per block.
    float* orow = out + (long long)(tile * 16 + rbase) * D + n;
    if (tile * 16 + 16 <= N) {
        #pragma unroll
        for (int r = 0; r < 8; ++r) orow[r * D] = acc[r] + bcol;
    } else {
        #pragma unroll
        for (int r = 0; r < 8; ++r) {
            if (tile * 16 + rbase + r < N) orow[r * D] = acc[r] + bcol;
        }
    }
}

// ---------------------------------------------------------------- launch
extern "C" void kernel_launch(void* const* d_in, const int* in_sizes, int n_in,
                              void* d_out, int out_size, void* d_ws, size_t ws_size,
                              hipStream_t stream) {
    const float* feature = (const float*)d_in[0];   // [N, 128]
    const float* Wm      = (const float*)d_in[1];   // [128, 128]
    const float* bias    = (const float*)d_in[2];   // [128]
    const int*   src     = (const int*)d_in[3];     // [E]
    const int*   dst     = (const int*)d_in[4];     // [E]

    float* out = (float*)d_out;
    float* deg = (float*)d_ws;                      // N floats of scratch

    const int N = in_sizes[0] / D;
    const int E = in_sizes[3];
    const int n_out = N * D;

    // 1) zero accumulator (d_out) + degree
    gcn_zero<<<4096, 256, 0, stream>>>(out, deg, n_out, N);

    // 2) edge scatter: E*32 threads
    long long scatter_threads = (long long)E * 32;
    int scatter_blocks = (int)((scatter_threads + 255) / 256);
    gcn_scatter<<<scatter_blocks, 256, 0, stream>>>(feature, src, dst, out, deg, E);

    // 3) mean + GEMM (WMMA f32), 16 nodes per block
    int gemm_blocks = (N + 15) / 16;
    gcn_gemm<<<gemm_blocks, 256, 0, stream>>>(Wm, bias, deg, out, N);
}